// APPNP_88321707475506
// MI455X (gfx1250) — compile-verified
//
#include <hip/hip_runtime.h>
#include <math.h>

// ---------------------------------------------------------------------------
// APPNP + edge attention for MI455X (gfx1250, wave32, WMMA)
// ---------------------------------------------------------------------------

#define N_NODES 50000
#define IN_DIM  512
#define HID_DIM 256
#define OUT_DIM 64
#define K_HOPS  4
#define ALPHA   0.1f

typedef __attribute__((ext_vector_type(2))) float v2f;
typedef __attribute__((ext_vector_type(8))) float v8f;

// ---------------------------------------------------------------------------
// Fused MLP:  feat0 = relu(h@W1 + b1) @ W2 + b2   (also copies into feat)
// One block (128 threads = 4 wave32) handles 16 rows.
// GEMM1: each wave owns 4 hidden col-tiles (4*16 = 64 cols; 4 waves = 256).
// GEMM2: each wave owns 1 output col-tile (16 cols; 4 waves = 64).
// Uses V_WMMA_F32_16X16X4_F32 (fp32-faithful; GEMMs are not the bottleneck).
// ---------------------------------------------------------------------------
__launch_bounds__(128)
__global__ void mlp_wmma_kernel(const float* __restrict__ h,
                                const float* __restrict__ W1,
                                const float* __restrict__ b1,
                                const float* __restrict__ W2,
                                const float* __restrict__ b2,
                                float* __restrict__ feat0,
                                float* __restrict__ feat) {
    // +4 float padding per row -> bank stride 4 (mod 64), no 16-way conflicts
    __shared__ float As[16][IN_DIM + 4];   // ~33 KB
    __shared__ float Hs[16][HID_DIM + 4];  // ~16.6 KB

    const int row0 = blockIdx.x * 16;
    const int tid  = threadIdx.x;
    const int wave = tid >> 5;
    const int lane = tid & 31;
    const int half = lane >> 4;   // lane half selects K pair (ISA 7.12.2)
    const int lm   = lane & 15;   // M (A-frag) / N (B,C,D-frags)

    // ---- Stage A tile: 16 rows x 512 cols = 2048 float4, 16 per thread ----
    {
        const float4* hv = (const float4*)h;
        #pragma unroll
        for (int i = 0; i < 16; ++i) {
            int idx = tid + i * 128;          // 0..2047
            int r   = idx >> 7;               // row 0..15
            int c   = idx & 127;              // float4 col 0..127
            float4 v = hv[(size_t)(row0 + r) * (IN_DIM / 4) + c];
            ((float4*)(&As[r][0]))[c] = v;
        }
    }
    __syncthreads();

    // -------------------- GEMM1: hidden = relu(A @ W1 + b1) ----------------
    v8f acc[4] = {};
    for (int k = 0; k < IN_DIM; k += 4) {
        const int krow = k + 2 * half;
        v2f a;
        a.x = As[lm][krow];
        a.y = As[lm][krow + 1];
        #pragma unroll
        for (int j = 0; j < 4; ++j) {
            const int n = (wave * 4 + j) * 16 + lm;
            v2f b;
            b.x = W1[(size_t)krow * HID_DIM + n];
            b.y = W1[(size_t)(krow + 1) * HID_DIM + n];
            acc[j] = __builtin_amdgcn_wmma_f32_16x16x4_f32(
                false, a, false, b, (short)0, acc[j], false, false);
        }
    }

    // bias + relu -> LDS hidden tile
    #pragma unroll
    for (int j = 0; j < 4; ++j) {
        const int n = (wave * 4 + j) * 16 + lm;
        const float bias = b1[n];
        #pragma unroll
        for (int r = 0; r < 8; ++r) {
            const int m = r + 8 * half;       // C/D layout: VGPR r -> row r+8*half
            float v = acc[j][r] + bias;
            Hs[m][n] = v > 0.0f ? v : 0.0f;
        }
    }
    __syncthreads();

    // -------------------- GEMM2: out = hidden @ W2 + b2 --------------------
    v8f oc = {};
    const int n = wave * 16 + lm;
    for (int k = 0; k < HID_DIM; k += 4) {
        const int krow = k + 2 * half;
        v2f a;
        a.x = Hs[lm][krow];
        a.y = Hs[lm][krow + 1];
        v2f b;
        b.x = W2[(size_t)krow * OUT_DIM + n];
        b.y = W2[(size_t)(krow + 1) * OUT_DIM + n];
        oc = __builtin_amdgcn_wmma_f32_16x16x4_f32(
            false, a, false, b, (short)0, oc, false, false);
    }
    const float bias = b2[n];
    #pragma unroll
    for (int r = 0; r < 8; ++r) {
        const int m = r + 8 * half;
        const float v = oc[r] + bias;
        const size_t o = (size_t)(row0 + m) * OUT_DIM + n;
        feat0[o] = v;
        feat[o]  = v;
    }
}

// ---------------------------------------------------------------------------
// Utility / graph kernels
// ---------------------------------------------------------------------------
__global__ void fill_kernel(float* __restrict__ p, float v, int n) {
    int i = blockIdx.x * blockDim.x + threadIdx.x;
    if (i < n) p[i] = v;
}

__global__ void degree_kernel(const int* __restrict__ src,
                              const int* __restrict__ dst,
                              float* __restrict__ deg_out,
                              float* __restrict__ deg_in, int E) {
    int e = blockIdx.x * blockDim.x + threadIdx.x;
    if (e >= E) return;
    atomicAdd(&deg_out[src[e]], 1.0f);
    atomicAdd(&deg_in[dst[e]], 1.0f);
}

__global__ void norm_kernel(const float* __restrict__ deg_o,
                            const float* __restrict__ deg_i,
                            float* __restrict__ ns,
                            float* __restrict__ nd, int N) {
    int i = blockIdx.x * blockDim.x + threadIdx.x;
    if (i >= N) return;
    ns[i] = rsqrtf(fmaxf(deg_o[i], 1.0f));
    nd[i] = rsqrtf(fmaxf(deg_i[i], 1.0f));
}

// One wave32 per edge; each lane handles a float2 slice (32*2 = 64 dims).
// feat/agg (12.8 MB each) are L2-resident on MI455X (192 MB L2).
__global__ void scatter_kernel(const float* __restrict__ feat,
                               const float* __restrict__ norm_src,
                               const int* __restrict__ src,
                               const int* __restrict__ dst,
                               float* __restrict__ agg, int E) {
    unsigned tid = blockIdx.x * blockDim.x + threadIdx.x;
    int e = tid >> 5;
    if (e >= E) return;
    int lane = tid & 31;
    int s = src[e], d = dst[e];
    float ns = norm_src[s];
    float2 f = ((const float2*)(feat + (size_t)s * OUT_DIM))[lane];
    float* out = agg + (size_t)d * OUT_DIM + lane * 2;
    atomicAdd(out + 0, f.x * ns);
    atomicAdd(out + 1, f.y * ns);
}

__global__ void combine_kernel(float* __restrict__ feat,
                               const float* __restrict__ agg,
                               const float* __restrict__ norm_dst,
                               const float* __restrict__ feat0, int n) {
    int i = blockIdx.x * blockDim.x + threadIdx.x;
    if (i >= n) return;
    float nd = norm_dst[i >> 6];  // OUT_DIM == 64
    feat[i] = (1.0f - ALPHA) * agg[i] * nd + ALPHA * feat0[i];
}

__global__ void edge_scores_kernel(const float* __restrict__ feat,
                                   const float* __restrict__ w_src,
                                   const float* __restrict__ w_dst,
                                   float* __restrict__ e_src,
                                   float* __restrict__ e_dst, int N) {
    int i = blockIdx.x * blockDim.x + threadIdx.x;
    if (i >= N) return;
    const float* x = feat + (size_t)i * OUT_DIM;
    float s1 = 0.0f, s2 = 0.0f;
    #pragma unroll
    for (int d = 0; d < OUT_DIM; ++d) {
        float v = x[d];
        s1 += v * w_src[d];
        s2 += v * w_dst[d];
    }
    e_src[i] = s1;
    e_dst[i] = s2;
}

__device__ inline void atomicMaxF(float* addr, float val) {
    // sign-aware int-punned float max (values here are tanh outputs, no NaN)
    if (val >= 0.0f)
        atomicMax((int*)addr, __float_as_int(val));
    else
        atomicMin((unsigned int*)addr, (unsigned int)__float_as_int(val));
}

__global__ void edge_tanh_max_kernel(const float* __restrict__ e_src,
                                     const float* __restrict__ e_dst,
                                     const int* __restrict__ src,
                                     const int* __restrict__ dst,
                                     float* __restrict__ ebuf,
                                     float* __restrict__ emax, int E) {
    int e = blockIdx.x * blockDim.x + threadIdx.x;
    if (e >= E) return;
    float v = tanhf(e_src[src[e]] + e_dst[dst[e]]);
    ebuf[e] = v;
    atomicMaxF(&emax[dst[e]], v);
}

__global__ void edge_exp_sum_kernel(float* __restrict__ ebuf,
                                    const float* __restrict__ emax,
                                    const int* __restrict__ dst,
                                    float* __restrict__ esum, int E) {
    int e = blockIdx.x * blockDim.x + threadIdx.x;
    if (e >= E) return;
    float ex = expf(ebuf[e] - emax[dst[e]]);
    ebuf[e] = ex;
    atomicAdd(&esum[dst[e]], ex);
}

__global__ void edge_attn_kernel(const float* __restrict__ ebuf,
                                 const float* __restrict__ esum,
                                 const int* __restrict__ dst,
                                 float* __restrict__ attn, int E) {
    int e = blockIdx.x * blockDim.x + threadIdx.x;
    if (e >= E) return;
    attn[e] = ebuf[e] / esum[dst[e]];
}

__global__ void log_softmax_kernel(const float* __restrict__ feat,
                                   float* __restrict__ out, int N) {
    int i = blockIdx.x * blockDim.x + threadIdx.x;
    if (i >= N) return;
    const float* x = feat + (size_t)i * OUT_DIM;
    float m = x[0];
    #pragma unroll
    for (int d = 1; d < OUT_DIM; ++d) m = fmaxf(m, x[d]);
    float s = 0.0f;
    #pragma unroll
    for (int d = 0; d < OUT_DIM; ++d) s += expf(x[d] - m);
    float ls = logf(s);
    float* o = out + (size_t)i * OUT_DIM;
    #pragma unroll
    for (int d = 0; d < OUT_DIM; ++d) o[d] = x[d] - m - ls;
}

// ---------------------------------------------------------------------------
// Launch
// ---------------------------------------------------------------------------
extern "C" void kernel_launch(void* const* d_in, const int* in_sizes, int n_in,
                              void* d_out, int out_size, void* d_ws, size_t ws_size,
                              hipStream_t stream) {
    const float* h     = (const float*)d_in[0];
    const int*   src   = (const int*)d_in[1];
    const int*   dst   = (const int*)d_in[2];
    const float* W1    = (const float*)d_in[3];
    const float* b1    = (const float*)d_in[4];
    const float* W2    = (const float*)d_in[5];
    const float* b2    = (const float*)d_in[6];
    const float* w_src = (const float*)d_in[7];
    const float* w_dst = (const float*)d_in[8];

    const int N = N_NODES;
    const int E = in_sizes[1];

    // workspace layout (floats)
    float* ws       = (float*)d_ws;
    float* deg_out  = ws; ws += N;
    float* deg_in   = ws; ws += N;
    float* norm_src = ws; ws += N;
    float* norm_dst = ws; ws += N;
    float* feat0    = ws; ws += (size_t)N * OUT_DIM;
    float* feat     = ws; ws += (size_t)N * OUT_DIM;
    float* agg      = ws; ws += (size_t)N * OUT_DIM;
    float* e_src    = ws; ws += N;
    float* e_dst    = ws; ws += N;
    float* e_max    = ws; ws += N;
    float* e_sum    = ws; ws += N;
    float* e_buf    = ws; ws += E;

    float* out_logsm = (float*)d_out;                         // N*64 floats
    float* out_attn  = (float*)d_out + (size_t)N * OUT_DIM;   // E floats

    const int BT = 256;
    auto blocks = [](long long n, int bt) { return (int)((n + bt - 1) / bt); };

    // degrees + norms
    fill_kernel<<<blocks(N, BT), BT, 0, stream>>>(deg_out, 0.0f, N);
    fill_kernel<<<blocks(N, BT), BT, 0, stream>>>(deg_in, 0.0f, N);
    degree_kernel<<<blocks(E, BT), BT, 0, stream>>>(src, dst, deg_out, deg_in, E);
    norm_kernel<<<blocks(N, BT), BT, 0, stream>>>(deg_out, deg_in, norm_src, norm_dst, N);

    // fused MLP (WMMA fp32): feat0 = MLP(h); feat = feat0
    mlp_wmma_kernel<<<N / 16, 128, 0, stream>>>(h, W1, b1, W2, b2, feat0, feat);

    // APPNP propagation (entirely L2-resident on MI455X)
    const int nf = N * OUT_DIM;
    for (int hop = 0; hop < K_HOPS; ++hop) {
        fill_kernel<<<blocks(nf, BT), BT, 0, stream>>>(agg, 0.0f, nf);
        scatter_kernel<<<blocks((long long)E * 32, BT), BT, 0, stream>>>(
            feat, norm_src, src, dst, agg, E);
        combine_kernel<<<blocks(nf, BT), BT, 0, stream>>>(feat, agg, norm_dst, feat0, nf);
    }

    // edge attention softmax
    edge_scores_kernel<<<blocks(N, BT), BT, 0, stream>>>(feat, w_src, w_dst, e_src, e_dst, N);
    fill_kernel<<<blocks(N, BT), BT, 0, stream>>>(e_max, -INFINITY, N);
    fill_kernel<<<blocks(N, BT), BT, 0, stream>>>(e_sum, 0.0f, N);
    edge_tanh_max_kernel<<<blocks(E, BT), BT, 0, stream>>>(e_src, e_dst, src, dst, e_buf, e_max, E);
    edge_exp_sum_kernel<<<blocks(E, BT), BT, 0, stream>>>(e_buf, e_max, dst, e_sum, E);
    edge_attn_kernel<<<blocks(E, BT), BT, 0, stream>>>(e_buf, e_sum, dst, out_attn, E);

    // output 1: row-wise log-softmax of feat
    log_softmax_kernel<<<blocks(N, BT), BT, 0, stream>>>(feat, out_logsm, N);
}